// Featurizer_71219147702438
// MI455X (gfx1250) — compile-verified
//
#include <hip/hip_runtime.h>
#include <hip/hip_bf16.h>

typedef __attribute__((ext_vector_type(2))) float v2f;
typedef __attribute__((ext_vector_type(8))) float v8f;

#define NN 256
#define NFEAT 24
#define RC_F 6.0f
#define PI_OVER_RC 0.5235987755982988f  // pi/6

// ---------------------------------------------------------------------------
// Kernel 1: one block per (b, j); thread = k.
// Builds pair tables P_eta[b][j][k] = fc(r_jk) * exp(-eta * sq_jk)  (diag = 0)
// and writes radial features out[b][j][0..7].
// ---------------------------------------------------------------------------
__global__ void __launch_bounds__(NN)
acsf_radial_pairs(const float* __restrict__ an,
                  const float* __restrict__ coord,
                  float* __restrict__ out,
                  float* __restrict__ P01,
                  float* __restrict__ P10) {
    const int j = blockIdx.x;
    const int b = blockIdx.y;
    const int k = threadIdx.x;

    __shared__ float red[NN];

    const float* cb = coord + (size_t)b * NN * 3;
    const float xj = cb[j * 3 + 0], yj = cb[j * 3 + 1], zj = cb[j * 3 + 2];
    const float xk = cb[k * 3 + 0], yk = cb[k * 3 + 1], zk = cb[k * 3 + 2];

    const float dx = xj - xk, dy = yj - yk, dz = zj - zk;
    const float sq = dx * dx + dy * dy + dz * dz;
    const float r  = sqrtf(sq);
    const bool  ok = (k != j) && (r < RC_F);
    const float fcr = ok ? 0.5f * (__cosf(r * PI_OVER_RC) + 1.0f) : 0.0f;

    // pair tables for the angular kernel
    const size_t base = ((size_t)b * NN + j) * NN + k;
    P01[base] = fcr * __expf(-0.01f * sq);
    P10[base] = fcr * __expf(-0.10f * sq);

    // radial per-thread partials: [fc, G2 x4, G3 x2]
    float part[7];
    part[0] = fcr;
    {
        const float d0 = r - 0.0f, d2 = r - 2.0f;
        part[1] = __expf(-0.1f * d0 * d0) * fcr;
        part[2] = __expf(-1.0f * d0 * d0) * fcr;
        part[3] = __expf(-0.1f * d2 * d2) * fcr;
        part[4] = __expf(-1.0f * d2 * d2) * fcr;
    }
    part[5] = __cosf(1.0f * r) * fcr;
    part[6] = __cosf(2.0f * r) * fcr;

    float* orow = out + ((size_t)b * NN + j) * NFEAT;
#pragma unroll
    for (int f = 0; f < 7; ++f) {
        red[k] = part[f];
        __syncthreads();
        for (int s = NN / 2; s > 0; s >>= 1) {
            if (k < s) red[k] += red[k + s];
            __syncthreads();
        }
        if (k == 0) orow[1 + f] = red[0];
        __syncthreads();
    }
    if (k == 0) orow[0] = an[j];
}

// ---------------------------------------------------------------------------
// Kernel 2: one block per (b, i); 256 threads = 8 waves.
// cos(theta_jik) tiles via V_WMMA_F32_16X16X4_F32 (Gram matrix of unit vecs),
// then per-lane polynomial accumulation of the 16 angular features.
// ---------------------------------------------------------------------------
__global__ void __launch_bounds__(NN)
acsf_angular(const float* __restrict__ coord,
             const float* __restrict__ P01,
             const float* __restrict__ P10,
             float* __restrict__ out) {
    const int i = blockIdx.x;
    const int b = blockIdx.y;
    const int tid = threadIdx.x;

    __shared__ float4 Us[NN];     // unit vectors (x,y,z,0) of dvec[i][j]
    __shared__ float  A01s[NN];   // fc_ij * exp(-0.01 sq_ij)
    __shared__ float  A10s[NN];   // fc_ij * exp(-0.10 sq_ij)
    __shared__ float  red[NN];

    // ---- per-(i,j) setup: thread = j ----
    {
        const float* cb = coord + (size_t)b * NN * 3;
        const float xi = cb[i * 3 + 0], yi = cb[i * 3 + 1], zi = cb[i * 3 + 2];
        const float dx = cb[tid * 3 + 0] - xi;
        const float dy = cb[tid * 3 + 1] - yi;
        const float dz = cb[tid * 3 + 2] - zi;
        const float sq = dx * dx + dy * dy + dz * dz;
        const float r  = sqrtf(sq);
        const bool  ok = (tid != i) && (r < RC_F);
        const float fcr = ok ? 0.5f * (__cosf(r * PI_OVER_RC) + 1.0f) : 0.0f;
        const float rinv = (sq > 0.0f) ? rsqrtf(sq) : 0.0f;
        Us[tid]  = make_float4(dx * rinv, dy * rinv, dz * rinv, 0.0f);
        A01s[tid] = fcr * __expf(-0.01f * sq);
        A10s[tid] = fcr * __expf(-0.10f * sq);
    }
    __syncthreads();

    const int lane = tid & 31;
    const int wave = tid >> 5;
    const int half = lane >> 4;   // 0: lanes 0-15, 1: lanes 16-31
    const int m    = lane & 15;

    const float* __restrict__ Pb01 = P01 + (size_t)b * NN * NN;
    const float* __restrict__ Pb10 = P10 + (size_t)b * NN * NN;

    float s4[8] = {0, 0, 0, 0, 0, 0, 0, 0};  // [eta*4 + zl], fc3/sq3 set
    float s5[8] = {0, 0, 0, 0, 0, 0, 0, 0};  // fc2/sq2 set

    // 256 16x16 tiles of the (j,k) plane; wave-uniform loop -> EXEC all ones.
    for (int t = wave; t < 256; t += 8) {
        const int J  = t >> 4;
        const int Kt = t & 15;

        // A: 16x4 (M=j-row, K=xyz0). lanes 0-15: K=0/1, lanes 16-31: K=2/3.
        const float4 uj = Us[(J << 4) + m];
        // B: 4x16 (K x N=k-col). v0: K=0 (lo) / K=1 (hi); v1: K=2 (lo) / K=3 (hi).
        const float4 uk = Us[(Kt << 4) + m];

        float cf[8];
#if defined(__has_builtin) && __has_builtin(__builtin_amdgcn_wmma_f32_16x16x4_f32)
        {
            v2f a, bm;
            a.x  = half ? uj.z : uj.x;
            a.y  = half ? uj.w : uj.y;   // K=3 row is zero padding
            bm.x = half ? uk.y : uk.x;
            bm.y = half ? uk.w : uk.z;   // K=3 row is zero padding
            v8f cacc = {0.f, 0.f, 0.f, 0.f, 0.f, 0.f, 0.f, 0.f};
            cacc = __builtin_amdgcn_wmma_f32_16x16x4_f32(
                false, a, false, bm, (short)0, cacc, false, false);
#pragma unroll
            for (int r = 0; r < 8; ++r) cf[r] = cacc[r];
        }
#else
        {
            const float4 ukf = Us[(Kt << 4) + m];
#pragma unroll
            for (int r = 0; r < 8; ++r) {
                const float4 u = Us[(J << 4) + (half << 3) + r];
                cf[r] = u.x * ukf.x + u.y * ukf.y + u.z * ukf.z;
            }
        }
#endif
        // C/D layout: VGPR r holds (M = r + 8*half, N = m) of the 16x16 tile.
        const int k    = (Kt << 4) + m;
        const float ak01 = A01s[k];
        const float ak10 = A10s[k];
        const int jbase  = (J << 4) + (half << 3);

#pragma unroll
        for (int r = 0; r < 8; ++r) {
            const int j = jbase + r;
            const float c   = cf[r];
            const float tp  = fmaxf(1.0f + c, 0.0f);
            const float tm  = fmaxf(1.0f - c, 0.0f);
            const float tp2 = tp * tp, tm2 = tm * tm;
            const float tp4 = tp2 * tp2, tm4 = tm2 * tm2;

            const float w01 = A01s[j] * ak01;
            const float w10 = A10s[j] * ak10;
            const float q01 = w01 * Pb01[j * NN + k];
            const float q10 = w10 * Pb10[j * NN + k];

            // param order per eta: (z=1,l=-1),(z=1,l=+1),(z=4,l=-1),(z=4,l=+1)
            s4[0] += tm * q01;  s4[1] += tp * q01;
            s4[2] += tm4 * q01; s4[3] += tp4 * q01;
            s4[4] += tm * q10;  s4[5] += tp * q10;
            s4[6] += tm4 * q10; s4[7] += tp4 * q10;
            if (j != k) {  // pair_mask[j,k] for the fc2 set
                s5[0] += tm * w01;  s5[1] += tp * w01;
                s5[2] += tm4 * w01; s5[3] += tp4 * w01;
                s5[4] += tm * w10;  s5[5] += tp * w10;
                s5[6] += tm4 * w10; s5[7] += tp4 * w10;
            }
        }
    }
    __syncthreads();

    // ---- block reduction of the 16 angular features ----
    float vals[16];
#pragma unroll
    for (int f = 0; f < 8; ++f) { vals[f] = s4[f]; vals[8 + f] = s5[f]; }

    float* orow = out + ((size_t)b * NN + i) * NFEAT;
#pragma unroll 1
    for (int f = 0; f < 16; ++f) {
        red[tid] = vals[f];
        __syncthreads();
        for (int s = NN / 2; s > 0; s >>= 1) {
            if (tid < s) red[tid] += red[tid + s];
            __syncthreads();
        }
        if (tid == 0) {
            const int zl = f & 3;
            const float scale = (zl >= 2) ? 0.125f : 1.0f;  // 2^(1-zeta)
            orow[8 + f] = scale * red[0];
        }
        __syncthreads();
    }
}

// ---------------------------------------------------------------------------
extern "C" void kernel_launch(void* const* d_in, const int* in_sizes, int n_in,
                              void* d_out, int out_size, void* d_ws, size_t ws_size,
                              hipStream_t stream) {
    const float* an    = (const float*)d_in[0];   // atomic_numbers (N)
    const float* coord = (const float*)d_in[1];   // coordinates (B,N,3)
    float* out = (float*)d_out;

    const int N = in_sizes[0];                    // 256
    const int B = in_sizes[1] / (3 * N);          // 2

    float* P01 = (float*)d_ws;                    // B*N*N floats
    float* P10 = P01 + (size_t)B * N * N;         // B*N*N floats (1 MB total)

    dim3 grid((unsigned)N, (unsigned)B);
    acsf_radial_pairs<<<grid, NN, 0, stream>>>(an, coord, out, P01, P10);
    acsf_angular<<<grid, NN, 0, stream>>>(coord, P01, P10, out);
}